// NeuralMemory_24343874634157
// MI455X (gfx1250) — compile-verified
//
#include <hip/hip_runtime.h>
#include <math.h>

// ---------------- problem constants ----------------
constexpr int B_    = 4;
constexpr int N_    = 4096;
constexpr int D_    = 128;
constexpr int HID_  = 512;
constexpr int CHUNK_= 64;
constexpr int NC_   = N_ / CHUNK_;   // 64
constexpr int BH_   = B_;            // HEADS == 1
constexpr int NTOK_ = N_;            // tokens per bh
constexpr int KB_   = 128;           // K-block for LDS-staged reduction GEMMs

typedef __attribute__((ext_vector_type(2))) float v2f;
typedef __attribute__((ext_vector_type(8))) float v8f;

// ---------------- WMMA fp32 16x16x4 ----------------
__device__ __forceinline__ v8f wmma4(v2f a, v2f b, v8f c) {
    // 8 args: (neg_a, A, neg_b, B, c_mod, C, reuse_a, reuse_b)
    return __builtin_amdgcn_wmma_f32_16x16x4_f32(
        false, a, false, b, (short)0, c, false, false);
}

// One wave accumulates into a 16x16 f32 tile, K in steps of 4.
// A layout: lanes 0-15 hold row m=lane, VGPR0=K+0, VGPR1=K+1; lanes 16-31: K+2,K+3.
// B layout: VGPR0: lanes0-15 row K+0, lanes16-31 row K+2; VGPR1: K+1 / K+3.
template <typename FA, typename FB>
__device__ __forceinline__ v8f gemm_tile_acc(v8f c, int K, FA loadA, FB loadB) {
    const int lane = threadIdx.x & 31;
    const int hf   = lane >> 4;   // 0 or 1
    const int l    = lane & 15;
    for (int k = 0; k < K; k += 4) {
        v2f a, b;
        a.x = loadA(l, k + 2 * hf + 0);
        a.y = loadA(l, k + 2 * hf + 1);
        b.x = loadB(k + 2 * hf + 0, l);
        b.y = loadB(k + 2 * hf + 1, l);
        c = wmma4(a, b, c);
    }
    return c;
}
template <typename FA, typename FB>
__device__ __forceinline__ v8f gemm_tile(int K, FA loadA, FB loadB) {
    v8f c = {};
    return gemm_tile_acc(c, K, loadA, loadB);
}

// C/D layout: VGPR r: lanes0-15 -> M=r, N=lane; lanes16-31 -> M=r+8, N=lane-16.
template <typename FS>
__device__ __forceinline__ void store_tile(v8f c, FS store) {
    const int lane = threadIdx.x & 31;
    const int hf   = lane >> 4;
    const int l    = lane & 15;
#pragma unroll
    for (int r = 0; r < 8; ++r) store(r + 8 * hf, l, c[r]);
}

// ---------------- math helpers (jax.nn.gelu tanh approximation) ----------------
__device__ __forceinline__ float gelu_f(float x) {
    const float c = 0.7978845608028654f; // sqrt(2/pi)
    float t = tanhf(c * (x + 0.044715f * x * x * x));
    return 0.5f * x * (1.0f + t);
}
__device__ __forceinline__ float gelu_grad(float x) {
    const float c = 0.7978845608028654f;
    float x2 = x * x;
    float t  = tanhf(c * (x + 0.044715f * x * x2));
    float sech2 = 1.0f - t * t;
    return 0.5f * (1.0f + t) + 0.5f * x * sech2 * c * (1.0f + 3.0f * 0.044715f * x2);
}
__device__ __forceinline__ float sigmoid_f(float x) { return 1.0f / (1.0f + expf(-x)); }

// ---------------- kernels ----------------
__global__ void zero_kernel(float* p, int n) {
    int i = blockIdx.x * blockDim.x + threadIdx.x;
    if (i < n) p[i] = 0.0f;
}

// k/v/q projections: (BH*N,128) @ (128,128); grid.z selects matrix.
__global__ __launch_bounds__(128) void proj_kernel(
    const float* __restrict__ seq,
    const float* __restrict__ wk, const float* __restrict__ wv, const float* __restrict__ wq,
    float* __restrict__ kbuf, float* __restrict__ vbuf, float* __restrict__ qbuf) {
    const int wave = threadIdx.x >> 5;                 // 4 waves
    const int row0 = blockIdx.x * 16;                  // 1024 row tiles
    const int col0 = (blockIdx.y * 4 + wave) * 16;     // 8 col tiles
    const float* W = (blockIdx.z == 0) ? wk : (blockIdx.z == 1) ? wv : wq;
    float* O       = (blockIdx.z == 0) ? kbuf : (blockIdx.z == 1) ? vbuf : qbuf;
    v8f acc = gemm_tile(D_,
        [&](int m, int kk) { return seq[(size_t)(row0 + m) * D_ + kk]; },
        [&](int kk, int n) { return W[(size_t)kk * D_ + col0 + n]; });
    store_tile(acc, [&](int m, int n, float v) {
        O[(size_t)(row0 + m) * D_ + col0 + n] = v;
    });
}

// per-(bh,chunk) data-dependent lr / decay(keep) / momentum(eta) from chunk's first token
__global__ __launch_bounds__(128) void hyper_kernel(
    const float* __restrict__ seq,
    const float* __restrict__ w_lr, const float* __restrict__ b_lr,
    const float* __restrict__ w_decay, const float* __restrict__ b_decay,
    const float* __restrict__ w_mom, const float* __restrict__ b_mom,
    float* __restrict__ lr, float* __restrict__ eta, float* __restrict__ keep) {
    const int bc = blockIdx.x;
    const int b = bc / NC_, cidx = bc % NC_;
    const float* rep = seq + ((size_t)b * N_ + (size_t)cidx * CHUNK_) * D_;
    const int t = threadIdx.x;
    float x = rep[t];
    __shared__ float s0[128], s1[128], s2[128];
    s0[t] = x * w_lr[t]; s1[t] = x * w_decay[t]; s2[t] = x * w_mom[t];
    __syncthreads();
    for (int off = 64; off > 0; off >>= 1) {
        if (t < off) { s0[t] += s0[t + off]; s1[t] += s1[t + off]; s2[t] += s2[t + off]; }
        __syncthreads();
    }
    if (t == 0) {
        lr[bc]   = sigmoid_f(s0[0] + b_lr[0]);
        keep[bc] = 1.0f - sigmoid_f(s1[0] + b_decay[0]);
        eta[bc]  = sigmoid_f(s2[0] + b_mom[0]);
    }
}

// collapse the two linear scans into per-chunk scalar coefficients:
// c_i = sum_{t>=i} (prod_{j>t} keep_j) * (prod_{j=i+1..t} eta_j)
__global__ __launch_bounds__(256) void coef_kernel(
    const float* __restrict__ eta, const float* __restrict__ keep, float* __restrict__ coef) {
    const int tid = threadIdx.x;
    const int bh = tid >> 6, i = tid & 63;
    __shared__ float Ak[BH_][NC_];
    if (i == 0) {
        float p = 1.0f;
        for (int t = NC_ - 1; t >= 0; --t) { Ak[bh][t] = p; p *= keep[bh * NC_ + t]; }
    }
    __syncthreads();
    float c = 0.0f, p = 1.0f;
    for (int t = i; t < NC_; ++t) {
        if (t > i) p *= eta[bh * NC_ + t];
        c += p * Ak[bh][t];
    }
    coef[tid] = c;
}

// per-(bh,chunk): MLP forward + RMSNorm backward; emits A=gelu(u), DU, DH (all scaled
// by wc = -c_t*lr_t*2/D via dpred) and atomically accumulates the final gamma.
__global__ __launch_bounds__(256) void grad_kernel(
    const float* __restrict__ kbuf, const float* __restrict__ vbuf,
    const float* __restrict__ w1_0, const float* __restrict__ w2_0,
    const float* __restrict__ gamma0,
    const float* __restrict__ lr, const float* __restrict__ coef,
    float* __restrict__ Abuf, float* __restrict__ DUbuf, float* __restrict__ DHbuf,
    float* __restrict__ gamma_f) {
    __shared__ float us[CHUNK_][HID_];   // 128 KB: pre-activation u
    __shared__ float hs[CHUNK_][D_];     //  32 KB
    __shared__ float dhs[CHUNK_][D_];    //  32 KB
    const int bc = blockIdx.x;
    const int bh = bc / NC_, cidx = bc % NC_;
    const float* kc = kbuf + ((size_t)bh * NTOK_ + (size_t)cidx * CHUNK_) * D_;
    const float* vc = vbuf + ((size_t)bh * NTOK_ + (size_t)cidx * CHUNK_) * D_;
    float* Ac  = Abuf  + ((size_t)bh * NTOK_ + (size_t)cidx * CHUNK_) * HID_;
    float* DUc = DUbuf + ((size_t)bh * NTOK_ + (size_t)cidx * CHUNK_) * HID_;
    float* DHc = DHbuf + ((size_t)bh * NTOK_ + (size_t)cidx * CHUNK_) * D_;
    const int wave = threadIdx.x >> 5;  // 8 waves
    const int lane = threadIdx.x & 31;

    // u = k_c @ W1 ; also emit A = gelu(u)
    for (int t = wave; t < (CHUNK_ / 16) * (HID_ / 16); t += 8) {
        int r0 = (t / (HID_ / 16)) * 16, c0 = (t % (HID_ / 16)) * 16;
        v8f acc = gemm_tile(D_,
            [&](int m, int kk) { return kc[(size_t)(r0 + m) * D_ + kk]; },
            [&](int kk, int n) { return w1_0[(size_t)kk * HID_ + c0 + n]; });
        store_tile(acc, [&](int m, int n, float v) {
            us[r0 + m][c0 + n] = v;
            Ac[(size_t)(r0 + m) * HID_ + c0 + n] = gelu_f(v);
        });
    }
    __syncthreads();

    // h = gelu(u) @ W2
    for (int t = wave; t < (CHUNK_ / 16) * (D_ / 16); t += 8) {
        int r0 = (t / (D_ / 16)) * 16, c0 = (t % (D_ / 16)) * 16;
        v8f acc = gemm_tile(HID_,
            [&](int m, int kk) { return gelu_f(us[r0 + m][kk]); },
            [&](int kk, int n) { return w2_0[(size_t)kk * D_ + c0 + n]; });
        store_tile(acc, [&](int m, int n, float v) { hs[r0 + m][c0 + n] = v; });
    }
    __syncthreads();

    // RMSNorm fwd+bwd per token (one wave per 8 tokens, 4 elems per lane)
    const float wc = -coef[bc] * lr[bc] * (2.0f / (float)D_);
    float gacc[4] = {0.f, 0.f, 0.f, 0.f};
    for (int tk = wave * 8; tk < wave * 8 + 8; ++tk) {
        float h0[4], ss = 0.0f;
#pragma unroll
        for (int j = 0; j < 4; ++j) { float hv = hs[tk][lane + 32 * j]; h0[j] = hv; ss += hv * hv; }
        for (int off = 16; off > 0; off >>= 1) ss += __shfl_xor(ss, off, 32);
        float rinv = 1.0f / sqrtf(ss * (1.0f / (float)D_) + 1e-6f);
        float nh[4], dnh[4], sdot = 0.0f;
#pragma unroll
        for (int j = 0; j < 4; ++j) {
            int dd = lane + 32 * j;
            nh[j] = h0[j] * rinv;
            float pred = nh[j] * gamma0[dd] + kc[(size_t)tk * D_ + dd];
            float dp = wc * (pred - vc[(size_t)tk * D_ + dd]);
            gacc[j] += dp * nh[j];
            dnh[j] = dp * gamma0[dd];
            sdot += dnh[j] * nh[j];
        }
        for (int off = 16; off > 0; off >>= 1) sdot += __shfl_xor(sdot, off, 32);
#pragma unroll
        for (int j = 0; j < 4; ++j) {
            int dd = lane + 32 * j;
            float dh = (dnh[j] - nh[j] * sdot * (1.0f / (float)D_)) * rinv;
            dhs[tk][dd] = dh;
            DHc[(size_t)tk * D_ + dd] = dh;
        }
    }
#pragma unroll
    for (int j = 0; j < 4; ++j) atomicAdd(&gamma_f[bh * D_ + lane + 32 * j], gacc[j]);
    __syncthreads();

    // DU = (dh @ W2^T) * gelu'(u)
    for (int t = wave; t < (CHUNK_ / 16) * (HID_ / 16); t += 8) {
        int r0 = (t / (HID_ / 16)) * 16, c0 = (t % (HID_ / 16)) * 16;
        v8f acc = gemm_tile(D_,
            [&](int m, int kk) { return dhs[r0 + m][kk]; },
            [&](int kk, int n) { return w2_0[(size_t)(c0 + n) * D_ + kk]; }); // W2^T
        store_tile(acc, [&](int m, int n, float v) {
            DUc[(size_t)(r0 + m) * HID_ + c0 + n] = v * gelu_grad(us[r0 + m][c0 + n]);
        });
    }
}

// w1_f[bh] = K^T @ DU : (128 x 4096) @ (4096 x 512)
// All 4 waves of a block share the same 16 output rows -> stage the shared 16xKB_
// A-slab (K^T) in LDS with coalesced loads; prefetch the next B K-block.
__global__ __launch_bounds__(128) void gw1_kernel(
    const float* __restrict__ kbuf, const float* __restrict__ DUbuf, float* __restrict__ w1_f) {
    __shared__ float As[16][KB_ + 1];   // +1 pad: kill LDS bank conflicts
    const int bh = blockIdx.x >> 6;
    const int tgroup = blockIdx.x & 63;
    const int wave = threadIdx.x >> 5;
    const int t = tgroup * 4 + wave;                 // 256 tiles (8 x 32)
    const int r0 = (tgroup / 8) * 16;                // shared across the 4 waves
    const int c0 = (t % (HID_ / 16)) * 16;
    const float* K  = kbuf  + (size_t)bh * NTOK_ * D_;
    const float* DU = DUbuf + (size_t)bh * NTOK_ * HID_;
    const int m_ld = threadIdx.x & 15;               // row within slab
    const int k_ld = threadIdx.x >> 4;               // 0..7
    v8f acc = {};
    for (int kb = 0; kb < NTOK_; kb += KB_) {
        // coalesced stage: As[m][kk] = K^T[r0+m][kb+kk] = K[(kb+kk)*D + r0+m]
        for (int kk = k_ld; kk < KB_; kk += 8)
            As[m_ld][kk] = K[(size_t)(kb + kk) * D_ + r0 + m_ld];
        if (kb + KB_ < NTOK_)  // stream the next B block ahead of the matrix pipe
            __builtin_prefetch(&DU[(size_t)(kb + KB_ + (threadIdx.x & 31)) * HID_ + c0], 0, 1);
        __syncthreads();
        acc = gemm_tile_acc(acc, KB_,
            [&](int m, int kk) { return As[m][kk]; },
            [&](int kk, int n) { return DU[(size_t)(kb + kk) * HID_ + c0 + n]; });
        __syncthreads();
    }
    store_tile(acc, [&](int m, int n, float v) {
        w1_f[((size_t)bh * D_ + r0 + m) * HID_ + c0 + n] = v;
    });
}

// w2_f[bh] = A^T @ DH : (512 x 4096) @ (4096 x 128), same LDS-staging scheme
__global__ __launch_bounds__(128) void gw2_kernel(
    const float* __restrict__ Abuf, const float* __restrict__ DHbuf, float* __restrict__ w2_f) {
    __shared__ float As[16][KB_ + 1];
    const int bh = blockIdx.x >> 6;
    const int tgroup = blockIdx.x & 63;
    const int wave = threadIdx.x >> 5;
    const int t = tgroup * 4 + wave;                 // 256 tiles (32 x 8)
    const int r0 = (tgroup / 2) * 16;                // shared across the 4 waves
    const int c0 = (t % (D_ / 16)) * 16;
    const float* A  = Abuf  + (size_t)bh * NTOK_ * HID_;
    const float* DH = DHbuf + (size_t)bh * NTOK_ * D_;
    const int m_ld = threadIdx.x & 15;
    const int k_ld = threadIdx.x >> 4;
    v8f acc = {};
    for (int kb = 0; kb < NTOK_; kb += KB_) {
        for (int kk = k_ld; kk < KB_; kk += 8)
            As[m_ld][kk] = A[(size_t)(kb + kk) * HID_ + r0 + m_ld];
        if (kb + KB_ < NTOK_)
            __builtin_prefetch(&DH[(size_t)(kb + KB_ + (threadIdx.x & 31)) * D_ + c0], 0, 1);
        __syncthreads();
        acc = gemm_tile_acc(acc, KB_,
            [&](int m, int kk) { return As[m][kk]; },
            [&](int kk, int n) { return DH[(size_t)(kb + kk) * D_ + c0 + n]; });
        __syncthreads();
    }
    store_tile(acc, [&](int m, int n, float v) {
        w2_f[((size_t)bh * HID_ + r0 + m) * D_ + c0 + n] = v;
    });
}

// out = RMSNorm(gelu(q@W1f)@W2f)*gamma_f + q
__global__ __launch_bounds__(256) void retrieve_kernel(
    const float* __restrict__ qbuf, const float* __restrict__ w1_f,
    const float* __restrict__ w2_f, const float* __restrict__ gamma_f,
    float* __restrict__ out) {
    __shared__ float us[CHUNK_][HID_];   // 128 KB
    __shared__ float hs[CHUNK_][D_];     //  32 KB
    const int bc = blockIdx.x;
    const int bh = bc / NC_, cidx = bc % NC_;
    const float* qc = qbuf + ((size_t)bh * NTOK_ + (size_t)cidx * CHUNK_) * D_;
    float* oc = out + ((size_t)bh * NTOK_ + (size_t)cidx * CHUNK_) * D_;
    const float* W1 = w1_f + (size_t)bh * D_ * HID_;
    const float* W2 = w2_f + (size_t)bh * HID_ * D_;
    const float* G  = gamma_f + bh * D_;
    const int wave = threadIdx.x >> 5;
    const int lane = threadIdx.x & 31;

    for (int t = wave; t < (CHUNK_ / 16) * (HID_ / 16); t += 8) {
        int r0 = (t / (HID_ / 16)) * 16, c0 = (t % (HID_ / 16)) * 16;
        v8f acc = gemm_tile(D_,
            [&](int m, int kk) { return qc[(size_t)(r0 + m) * D_ + kk]; },
            [&](int kk, int n) { return W1[(size_t)kk * HID_ + c0 + n]; });
        store_tile(acc, [&](int m, int n, float v) { us[r0 + m][c0 + n] = v; });
    }
    __syncthreads();
    for (int t = wave; t < (CHUNK_ / 16) * (D_ / 16); t += 8) {
        int r0 = (t / (D_ / 16)) * 16, c0 = (t % (D_ / 16)) * 16;
        v8f acc = gemm_tile(HID_,
            [&](int m, int kk) { return gelu_f(us[r0 + m][kk]); },
            [&](int kk, int n) { return W2[(size_t)kk * D_ + c0 + n]; });
        store_tile(acc, [&](int m, int n, float v) { hs[r0 + m][c0 + n] = v; });
    }
    __syncthreads();
    for (int tk = wave * 8; tk < wave * 8 + 8; ++tk) {
        float h0[4], ss = 0.0f;
#pragma unroll
        for (int j = 0; j < 4; ++j) { float hv = hs[tk][lane + 32 * j]; h0[j] = hv; ss += hv * hv; }
        for (int off = 16; off > 0; off >>= 1) ss += __shfl_xor(ss, off, 32);
        float rinv = 1.0f / sqrtf(ss * (1.0f / (float)D_) + 1e-6f);
#pragma unroll
        for (int j = 0; j < 4; ++j) {
            int dd = lane + 32 * j;
            oc[(size_t)tk * D_ + dd] = h0[j] * rinv * G[dd] + qc[(size_t)tk * D_ + dd];
        }
    }
}

// ---------------- host launcher ----------------
extern "C" void kernel_launch(void* const* d_in, const int* in_sizes, int n_in,
                              void* d_out, int out_size, void* d_ws, size_t ws_size,
                              hipStream_t stream) {
    (void)in_sizes; (void)n_in; (void)out_size; (void)ws_size;
    const float* seq     = (const float*)d_in[0];
    const float* w_k     = (const float*)d_in[1];
    const float* w_v     = (const float*)d_in[2];
    const float* w_q     = (const float*)d_in[3];
    const float* w_lr    = (const float*)d_in[4];
    const float* b_lr    = (const float*)d_in[5];
    const float* w_decay = (const float*)d_in[6];
    const float* b_decay = (const float*)d_in[7];
    const float* w_mom   = (const float*)d_in[8];
    const float* b_mom   = (const float*)d_in[9];
    const float* gamma0  = (const float*)d_in[10];
    const float* w1_0    = (const float*)d_in[11];
    const float* w2_0    = (const float*)d_in[12];
    float* out = (float*)d_out;

    float* ws = (float*)d_ws;
    const size_t KV = (size_t)BH_ * NTOK_ * D_;     // 2,097,152 floats
    const size_t AH = (size_t)BH_ * NTOK_ * HID_;   // 8,388,608 floats
    float* kbuf  = ws; ws += KV;
    float* vbuf  = ws; ws += KV;
    float* qbuf  = ws; ws += KV;
    float* DHbuf = ws; ws += KV;
    float* Abuf  = ws; ws += AH;
    float* DUbuf = ws; ws += AH;
    float* lr    = ws; ws += BH_ * NC_;
    float* eta   = ws; ws += BH_ * NC_;
    float* keep  = ws; ws += BH_ * NC_;
    float* coef  = ws; ws += BH_ * NC_;
    float* gamma_f = ws; ws += BH_ * D_;
    float* w1_f  = ws; ws += (size_t)BH_ * D_ * HID_;
    float* w2_f  = ws; ws += (size_t)BH_ * HID_ * D_;

    zero_kernel<<<dim3(2), dim3(256), 0, stream>>>(gamma_f, BH_ * D_);
    proj_kernel<<<dim3((BH_ * NTOK_) / 16, 2, 3), dim3(128), 0, stream>>>(
        seq, w_k, w_v, w_q, kbuf, vbuf, qbuf);
    hyper_kernel<<<dim3(BH_ * NC_), dim3(128), 0, stream>>>(
        seq, w_lr, b_lr, w_decay, b_decay, w_mom, b_mom, lr, eta, keep);
    coef_kernel<<<dim3(1), dim3(256), 0, stream>>>(eta, keep, coef);
    grad_kernel<<<dim3(BH_ * NC_), dim3(256), 0, stream>>>(
        kbuf, vbuf, w1_0, w2_0, gamma0, lr, coef, Abuf, DUbuf, DHbuf, gamma_f);
    gw1_kernel<<<dim3(BH_ * 64), dim3(128), 0, stream>>>(kbuf, DUbuf, w1_f);
    gw2_kernel<<<dim3(BH_ * 64), dim3(128), 0, stream>>>(Abuf, DHbuf, w2_f);
    retrieve_kernel<<<dim3(BH_ * NC_), dim3(256), 0, stream>>>(
        qbuf, w1_f, w2_f, gamma_f, out);
}